// CatmullRomActivation_52956946760055
// MI455X (gfx1250) — compile-verified
//
#include <hip/hip_runtime.h>
#include <math.h>
#include <stdint.h>

typedef __attribute__((ext_vector_type(2))) float v2f;
typedef __attribute__((ext_vector_type(4))) float f32x4;
typedef __attribute__((ext_vector_type(8))) float v8f;
typedef __attribute__((ext_vector_type(4))) int   v4i;

// Problem sizes (fixed by the reference)
#define M_DIM 16384
#define D_DIM 1024
#define N_DIM 1024
#define KCP   34      // control points per neuron (incl. ghosts)

// Tiling
constexpr int BM = 128;          // rows of C per block
constexpr int BN = 64;           // cols of C per block
constexpr int BK = 32;           // K depth per stage
constexpr int LDA = BK + 4;      // 36 dwords: multiple of 4 (b128 alignment), conflict-friendly
constexpr int LDB = BN + 8;      // 72 dwords: rows k,k+2 land on disjoint bank halves
constexpr int NTHREADS = 256;    // 8 waves (wave32)

// Pointer types matching __builtin_amdgcn_global_load_async_to_lds_b128:
//   arg0: v4i in addrspace(1) (global), arg1: v4i in addrspace(3) (LDS)
typedef __attribute__((address_space(1))) v4i as1_v4i;
typedef __attribute__((address_space(3))) v4i as3_v4i;

__device__ __forceinline__ as1_v4i* to_global(const void* p) {
    return (as1_v4i*)(unsigned long long)(uintptr_t)p;
}
// Generic->AS3: hardware maps generic LDS addresses via addr[31:0]
__device__ __forceinline__ as3_v4i* to_lds(void* p) {
    return (as3_v4i*)(unsigned int)(uintptr_t)p;
}

__global__ __launch_bounds__(NTHREADS)
void CatmullRomActivation_kernel(const float* __restrict__ X,
                                 const float* __restrict__ W,
                                 const float* __restrict__ CP,
                                 float* __restrict__ out)
{
    __shared__ float lds_a[2][BM * LDA];   // 2 * 128*36*4 = 36864 B
    __shared__ float lds_b[2][BK * LDB];   // 2 *  32*72*4 = 18432 B
    __shared__ float lds_cp[BN * KCP];     //       64*34*4 =  8704 B

    const int tid   = threadIdx.x;
    const int lane  = tid & 31;
    const int wave  = tid >> 5;
    const int half  = lane >> 4;     // which 16-lane half of the wave
    const int ln    = lane & 15;

    const int waveM = wave & 3;      // 4 waves along M (32 rows each)
    const int waveN = wave >> 2;     // 2 waves along N (32 cols each)

    const int block_m = blockIdx.y * BM;
    const int block_n = blockIdx.x * BN;

    // ---- Preload this block's control-point rows (contiguous: stride == KCP) ----
    for (int i = tid; i < BN * KCP; i += NTHREADS) {
        lds_cp[i] = CP[(size_t)block_n * KCP + i];
    }

    // ---- Global -> LDS staging via CDNA5 async DMA (no VGPR round-trip) ----
    auto load_tiles_async = [&](int k0, int buf) {
        // A tile: BM x BK  (1024 x 16B transfers, 4 per thread)
        #pragma unroll
        for (int i = 0; i < 4; ++i) {
            int f4  = tid + i * NTHREADS;       // 0..1023
            int row = f4 >> 3;                  // / (BK/4)
            int c4  = (f4 & 7) << 2;
            const float* g = X + (size_t)(block_m + row) * D_DIM + k0 + c4;
            float*       l = &lds_a[buf][row * LDA + c4];
            __builtin_amdgcn_global_load_async_to_lds_b128(to_global(g), to_lds(l), 0, 0);
        }
        // B tile: BK x BN  (512 x 16B transfers, 2 per thread)
        #pragma unroll
        for (int i = 0; i < 2; ++i) {
            int f4  = tid + i * NTHREADS;       // 0..511
            int row = f4 >> 4;                  // / (BN/4)
            int c4  = (f4 & 15) << 2;
            const float* g = W + (size_t)(k0 + row) * N_DIM + block_n + c4;
            float*       l = &lds_b[buf][row * LDB + c4];
            __builtin_amdgcn_global_load_async_to_lds_b128(to_global(g), to_lds(l), 0, 0);
        }
    };

    v8f acc[2][2] = {};

    load_tiles_async(0, 0);
    asm volatile("s_wait_asynccnt 0x0" ::: "memory");
    __syncthreads();

    for (int k0 = 0; k0 < D_DIM; k0 += BK) {
        const int buf = (k0 / BK) & 1;
        if (k0 + BK < D_DIM) load_tiles_async(k0 + BK, buf ^ 1);

        const float* A = lds_a[buf];
        const float* B = lds_b[buf];

        #pragma unroll
        for (int kk = 0; kk < BK; kk += 4) {
            // A fragment layout (16x4 f32): lanes 0-15 hold K=kk..kk+1, lanes 16-31 K=kk+2..kk+3
            v2f afrag[2];
            #pragma unroll
            for (int i = 0; i < 2; ++i) {
                int row = waveM * 32 + i * 16 + ln;
                int col = kk + half * 2;
                afrag[i] = *reinterpret_cast<const v2f*>(&A[row * LDA + col]);  // ds_load_b64
            }
            // B fragment layout (4x16 f32): VGPR0 = rows kk / kk+2, VGPR1 = rows kk+1 / kk+3
            v2f bfrag[2];
            #pragma unroll
            for (int j = 0; j < 2; ++j) {
                int col = waveN * 32 + j * 16 + ln;
                int row = kk + half * 2;
                v2f b;
                b[0] = B[(row    ) * LDB + col];
                b[1] = B[(row + 1) * LDB + col];
                bfrag[j] = b;
            }
            #pragma unroll
            for (int i = 0; i < 2; ++i)
                #pragma unroll
                for (int j = 0; j < 2; ++j)
                    acc[i][j] = __builtin_amdgcn_wmma_f32_16x16x4_f32(
                        /*neg_a=*/false, afrag[i],
                        /*neg_b=*/false, bfrag[j],
                        /*c_mod=*/(short)0, acc[i][j],
                        /*reuse_a=*/false, /*reuse_b=*/false);
        }

        // All of this wave's async prefetches must land, then block-wide rendezvous
        asm volatile("s_wait_asynccnt 0x0" ::: "memory");
        __syncthreads();
    }

    // ---- Fused Catmull-Rom spline epilogue ----
    // C/D layout: VGPR e, lanes 0-15 -> row m0+e, lanes 16-31 -> row m0+8+e; col = n0+ln
    #pragma unroll
    for (int i = 0; i < 2; ++i) {
        #pragma unroll
        for (int j = 0; j < 2; ++j) {
            const int n_local = waveN * 32 + j * 16 + ln;
            const float* cpr  = &lds_cp[n_local * KCP];
            const int gcol    = block_n + n_local;
            #pragma unroll
            for (int e = 0; e < 8; ++e) {
                const int grow = block_m + waveM * 32 + i * 16 + half * 8 + e;
                const float s  = acc[i][j][e];

                // p0 = floor((s - RANGE_MIN)*(CP_NUM-2)/(RANGE_MAX-RANGE_MIN) + 1)
                //    = floor(3.75*s + 16), with boundary selects
                float p0f = floorf(fmaf(s, 3.75f, 16.0f));
                p0f = (s <= -4.0f) ? 1.0f  : p0f;
                p0f = (s >=  4.0f) ? 31.0f : p0f;
                const int p0 = (int)p0f;

                // u = frac(s / DELTA_X), DELTA_X = 0.25
                const float t  = s * 4.0f;
                const float u  = t - floorf(t);
                const float u2 = u * u;
                const float u3 = u2 * u;

                // U(u) @ BASIS  (Catmull-Rom, factor 0.5)
                const float b0 = 0.5f * (-u3 + 2.0f * u2 - u);
                const float b1 = 0.5f * ( 3.0f * u3 - 5.0f * u2 + 2.0f);
                const float b2 = 0.5f * (-3.0f * u3 + 4.0f * u2 + u);
                const float b3 = 0.5f * ( u3 - u2);

                const float q0 = cpr[p0 - 1];
                const float q1 = cpr[p0    ];
                const float q2 = cpr[p0 + 1];
                const float q3 = cpr[p0 + 2];

                out[(size_t)grow * N_DIM + gcol] =
                    b0 * q0 + b1 * q1 + b2 * q2 + b3 * q3;
            }
        }
    }
}

extern "C" void kernel_launch(void* const* d_in, const int* in_sizes, int n_in,
                              void* d_out, int out_size, void* d_ws, size_t ws_size,
                              hipStream_t stream) {
    const float* X  = (const float*)d_in[0];   // (M, D) fp32
    const float* W  = (const float*)d_in[1];   // (D, N) fp32
    const float* CP = (const float*)d_in[2];   // (N, KCP) fp32
    float* out      = (float*)d_out;           // (M, N) fp32

    (void)in_sizes; (void)n_in; (void)out_size; (void)d_ws; (void)ws_size;

    dim3 grid(N_DIM / BN, M_DIM / BM);         // (16, 128)
    dim3 block(NTHREADS);
    CatmullRomActivation_kernel<<<grid, block, 0, stream>>>(X, W, CP, out);
}